// DecodingLoss_16501264351911
// MI455X (gfx1250) — compile-verified
//
#include <hip/hip_runtime.h>
#include <hip/hip_bf16.h>
#include <cstdint>

#define WIN 1024
#define BETA 100.0f
#define CEPS 1e-5f
#define PI2 6.283185307179586476f

typedef __attribute__((ext_vector_type(2))) float v2f;
typedef __attribute__((ext_vector_type(8))) float v8f;

__device__ constexpr int rev5(int x) {
  return ((x & 1) << 4) | ((x & 2) << 2) | (x & 4) | ((x & 8) >> 2) | ((x & 16) >> 4);
}

// In-register 32-point complex DIT FFT (input pre-bit-reversed, output natural order).
// Twiddles W32^j = exp(-2*pi*i*j/32) as compile-time constants -> fold to immediates.
__device__ __forceinline__ void fft32(float re[32], float im[32]) {
  constexpr float TWR[16] = {
      1.0f,  0.98078528f,  0.92387953f,  0.83146961f,
      0.70710678f,  0.55557023f,  0.38268343f,  0.19509032f,
      0.0f, -0.19509032f, -0.38268343f, -0.55557023f,
     -0.70710678f, -0.83146961f, -0.92387953f, -0.98078528f};
  constexpr float TWI[16] = {
      0.0f, -0.19509032f, -0.38268343f, -0.55557023f,
     -0.70710678f, -0.83146961f, -0.92387953f, -0.98078528f,
     -1.0f, -0.98078528f, -0.92387953f, -0.83146961f,
     -0.70710678f, -0.55557023f, -0.38268343f, -0.19509032f};
#pragma unroll
  for (int s = 1; s <= 5; ++s) {
    const int m = 1 << s, half = m >> 1, step = 32 >> s;
#pragma unroll
    for (int k = 0; k < 32; k += m) {
#pragma unroll
      for (int j = 0; j < half; ++j) {
        const float wr = TWR[j * step], wi = TWI[j * step];
        const int a = k + j, b = a + half;
        const float tr = wr * re[b] - wi * im[b];
        const float ti = wr * im[b] + wi * re[b];
        re[b] = re[a] - tr; im[b] = im[a] - ti;
        re[a] += tr;        im[a] += ti;
      }
    }
  }
}

// 8 waves / block, 2 windows per wave -> 16 windows per block (one WMMA tile).
// Window data is staged global->LDS with async b128 copies (ASYNCcnt double-buffer);
// the stage buffer is then reused as the padded transpose buffer (DS in-order per wave).
__global__ __launch_bounds__(256) void decode_main(const float* __restrict__ audio,
                                                   const int* __restrict__ symbols,
                                                   float* __restrict__ perAudioErr,
                                                   int blocksPerAudio) {
  __shared__ float ldsT[8][2 * 1056];  // per-wave: two stage/transpose buffers (stride-33 pad)
  __shared__ float Smat[16][32];       // folded log-mag sums: row=window, col=k mod 32
  __shared__ float Dmat[16][16];       // WMMA output: row=window, col=delay index
  __shared__ float errbuf[16];

  const int tid = threadIdx.x;
  const int wv  = tid >> 5;            // wave id 0..7
  const int L   = tid & 31;            // lane id
  const int blockBase = blockIdx.x * 16;

  // ---- Issue async global->LDS copies for BOTH windows this wave owns. ----
  // 8 x b128 per window: each instruction moves 512B for the wave; INST_OFFSET
  // applies to both the global and LDS sides, so one base address pair suffices.
#pragma unroll
  for (int it = 0; it < 2; ++it) {
    const long long base = (long long)(blockBase + wv + 8 * it) * WIN;
    const uint32_t la = (uint32_t)(uintptr_t)(&ldsT[wv][it * 1056]) + L * 16;
    const uint64_t ga = (uint64_t)(uintptr_t)(audio + base) + (uint64_t)L * 16;
    asm volatile("global_load_async_to_lds_b128 %0, %1, off offset:0"    :: "v"(la), "v"(ga) : "memory");
    asm volatile("global_load_async_to_lds_b128 %0, %1, off offset:512"  :: "v"(la), "v"(ga) : "memory");
    asm volatile("global_load_async_to_lds_b128 %0, %1, off offset:1024" :: "v"(la), "v"(ga) : "memory");
    asm volatile("global_load_async_to_lds_b128 %0, %1, off offset:1536" :: "v"(la), "v"(ga) : "memory");
    asm volatile("global_load_async_to_lds_b128 %0, %1, off offset:2048" :: "v"(la), "v"(ga) : "memory");
    asm volatile("global_load_async_to_lds_b128 %0, %1, off offset:2560" :: "v"(la), "v"(ga) : "memory");
    asm volatile("global_load_async_to_lds_b128 %0, %1, off offset:3072" :: "v"(la), "v"(ga) : "memory");
    asm volatile("global_load_async_to_lds_b128 %0, %1, off offset:3584" :: "v"(la), "v"(ga) : "memory");
  }

#pragma unroll
  for (int it = 0; it < 2; ++it) {
    const int row = wv + 8 * it;                 // Smat row / window-in-block
    float* tbuf = &ldsT[wv][it * 1056];

    // Async dones return in order: <=8 outstanding means window 0 landed.
    if (it == 0) asm volatile("s_wait_asynccnt 0x8" ::: "memory");
    else         asm volatile("s_wait_asynccnt 0x0" ::: "memory");

    // Lane L = column n2 of the 32x32 view: x[n1*32 + L], read bit-reversed for DIT.
    // Fixed n1 -> consecutive lanes -> conflict-free LDS reads.
    float re[32], im[32];
#pragma unroll
    for (int n1 = 0; n1 < 32; ++n1) {
      re[rev5(n1)] = tbuf[n1 * 32 + L];
      im[rev5(n1)] = 0.0f;
    }
    fft32(re, im);                               // column FFT over n1

    // Twiddle element k1 by W_1024^{L*k1}: one sincos + incremental rotation.
    float stC, stS;
    __sincosf((-PI2 / 1024.0f) * (float)L, &stS, &stC);
    float curC = stC, curS = stS;
#pragma unroll
    for (int k1 = 1; k1 < 32; ++k1) {
      const float r = re[k1] * curC - im[k1] * curS;
      const float i = re[k1] * curS + im[k1] * curC;
      re[k1] = r; im[k1] = i;
      const float nC = curC * stC - curS * stS;
      const float nS = curC * stS + curS * stC;
      curC = nC; curS = nS;
    }

    // 32x32 transpose via padded LDS (stride 33, conflict-free both ways), re then im.
    // DS ops are in-order per wave; s_wait_dscnt is the VGPR-hazard + compiler fence.
#pragma unroll
    for (int k1 = 0; k1 < 32; ++k1) tbuf[k1 * 33 + L] = re[k1];
    asm volatile("s_wait_dscnt 0" ::: "memory");
#pragma unroll
    for (int n2 = 0; n2 < 32; ++n2) re[rev5(n2)] = tbuf[L * 33 + n2];
    asm volatile("s_wait_dscnt 0" ::: "memory");
#pragma unroll
    for (int k1 = 0; k1 < 32; ++k1) tbuf[k1 * 33 + L] = im[k1];
    asm volatile("s_wait_dscnt 0" ::: "memory");
#pragma unroll
    for (int n2 = 0; n2 < 32; ++n2) im[rev5(n2)] = tbuf[L * 33 + n2];
    asm volatile("s_wait_dscnt 0" ::: "memory");

    fft32(re, im);                               // row FFT -> lane L holds X[L+32j]

    // Folded sum S[L] = sum over k<=512, k mod 32 == L, of w_k*log(|X_k|+eps)
    // (w=2 except w=1 at k=0 and k=512; both live in lane 0).
    float sAcc = 0.0f;
#pragma unroll
    for (int j = 0; j < 16; ++j) {
      const float mag = sqrtf(re[j] * re[j] + im[j] * im[j]);
      sAcc += 2.0f * __logf(mag + CEPS);
    }
    if (L == 0) {
      const float mag16 = sqrtf(re[16] * re[16] + im[16] * im[16]);
      const float mag0  = sqrtf(re[0] * re[0] + im[0] * im[0]);
      sAcc += __logf(mag16 + CEPS) - __logf(mag0 + CEPS);
    }
    Smat[row][L] = sAcc;
  }
  __syncthreads();

  // Cosine projection as one 16x32 @ 32x16 GEMM on the matrix core:
  // D[t][c] = sum_r S_t[r] * cos(2*pi*r*(c+1)/32)   (cols 8..15 zero).
  // 8 chained v_wmma_f32_16x16x4_f32. Wave 0 only (EXEC all 1s inside the branch).
  if (tid < 32) {
    const int h = tid >> 4, n = tid & 15;   // n = A-row (window) and B-col (delay)

    // Hoist the 16 B-matrix values this lane contributes (removes per-WMMA exec dance).
    float bv[16];
    if (n < 8) {
      const float w = PI2 * (float)(n + 1) / 32.0f;
#pragma unroll
      for (int kk = 0; kk < 8; ++kk) {
        bv[2 * kk]     = __cosf(w * (float)(4 * kk + 2 * h));
        bv[2 * kk + 1] = __cosf(w * (float)(4 * kk + 2 * h + 1));
      }
    } else {
#pragma unroll
      for (int i = 0; i < 16; ++i) bv[i] = 0.0f;
    }

    v8f d = {0.f, 0.f, 0.f, 0.f, 0.f, 0.f, 0.f, 0.f};
#pragma unroll
    for (int kk = 0; kk < 8; ++kk) {
      const int r0 = 4 * kk + 2 * h;        // K-slice per A 16x4 / B 4x16 lane layout
      v2f a, b;
      a.x = Smat[n][r0];
      a.y = Smat[n][r0 + 1];
      b.x = bv[2 * kk];
      b.y = bv[2 * kk + 1];
      d = __builtin_amdgcn_wmma_f32_16x16x4_f32(false, a, false, b, (short)0, d,
                                                false, false);
    }
#pragma unroll
    for (int v = 0; v < 8; ++v) Dmat[v + 8 * h][n] = d[v];   // C/D layout: M=v(+8h), N=lane
  }
  __syncthreads();

  // Softargmax + clipped symbol error, one thread per window.
  if (tid < 16) {
    float vals[8], mx = -1e30f;
#pragma unroll
    for (int c = 0; c < 8; ++c) {
      vals[c] = Dmat[tid][c] * (BETA / 1024.0f);   // (1/1024) irfft scale * beta
      mx = fmaxf(mx, vals[c]);
    }
    float den = 0.0f, num = 0.0f;
#pragma unroll
    for (int c = 0; c < 8; ++c) {
      const float e = __expf(vals[c] - mx);
      den += e; num += e * (float)c;
    }
    const float dec = num / den;
    const float sym = (float)symbols[blockBase + tid];
    errbuf[tid] = fminf(fabsf(dec - sym), 1.0f);
  }
  __syncthreads();

  if (tid == 0) {
    float s = 0.0f;
#pragma unroll
    for (int t = 0; t < 16; ++t) s += errbuf[t];
    atomicAdd(&perAudioErr[blockIdx.x / blocksPerAudio], s);
  }
}

__global__ void zero_ws(float* ws, int n) {
  const int i = blockIdx.x * blockDim.x + threadIdx.x;
  if (i < n) ws[i] = 0.0f;
}

__global__ void decode_finalize(const float* __restrict__ perAudioErr,
                                const int* __restrict__ nrI,
                                const int* __restrict__ rvI,
                                float* __restrict__ out, int B, int winsPerAudio) {
  __shared__ float sInv[64], sTot[64], sNr[64], sRv[64];
  const int t = threadIdx.x;
  if (t < B) {
    const float pae = perAudioErr[t];
    const float nr = (float)nrI[t], rv = (float)rvI[t];
    const float denom = rv - nr;
    const float er = (denom == 0.0f) ? 1.0f : (denom - (pae - nr)) / denom;
    sInv[t] = 1.0f / er;
    sTot[t] = pae; sNr[t] = nr; sRv[t] = rv;
  }
  __syncthreads();
  if (t == 0) {
    float tot = 0.f, inv = 0.f, snr = 0.f, srv = 0.f;
    for (int i = 0; i < B; ++i) { tot += sTot[i]; inv += sInv[i]; snr += sNr[i]; srv += sRv[i]; }
    const float nTot = (float)B * (float)winsPerAudio;
    out[0] = tot / nTot;     // sym_err_rate
    out[1] = tot;            // tot_sym_err
    out[2] = inv / (float)B; // avg_err_reduction
    out[3] = snr / nTot;     // gt_err_rate_no_reverb
    out[4] = srv / nTot;     // gt_err_rate_reverb
  }
}

extern "C" void kernel_launch(void* const* d_in, const int* in_sizes, int n_in,
                              void* d_out, int out_size, void* d_ws, size_t ws_size,
                              hipStream_t stream) {
  const float* audio   = (const float*)d_in[0];
  const int*   symbols = (const int*)d_in[1];
  const int*   nr      = (const int*)d_in[2];
  const int*   rv      = (const int*)d_in[3];
  float* pae = (float*)d_ws;
  float* out = (float*)d_out;

  const int B = in_sizes[2];                               // 64
  const int winsPerAudio = (in_sizes[0] / B) / WIN;        // 1024
  const int numWindows = B * winsPerAudio;                 // 65536
  const int numBlocks = numWindows / 16;                   // 4096
  const int blocksPerAudio = winsPerAudio / 16;            // 64

  hipLaunchKernelGGL(zero_ws, dim3(1), dim3(64), 0, stream, pae, B);
  hipLaunchKernelGGL(decode_main, dim3(numBlocks), dim3(256), 0, stream,
                     audio, symbols, pae, blocksPerAudio);
  hipLaunchKernelGGL(decode_finalize, dim3(1), dim3(64), 0, stream,
                     pae, nr, rv, out, B, winsPerAudio);
}